// InputTransition_11106785428134
// MI455X (gfx1250) — compile-verified
//
#include <hip/hip_runtime.h>
#include <stdint.h>

// Problem geometry (fixed by the reference): x = [2,1,64,256,256] f32,
// out = [2,16,64,256,256] f32 with channels [sobel, x] * 8.
#define IMH 256
#define IMW 256
#define IMHW (IMH * IMW)
#define NDEPTH 64
#define NSLICE 128                 // B*D = 2*64
#define CH_STRIDE (NDEPTH * IMHW)  // output channel stride (elements)

#define TILE_H 16
#define TILE_W 64
#define LDS_H 18                   // TILE_H + 2 halo rows
#define LDS_W 72                   // padded row stride (floats) = 288B, 16B-aligned
// LDS columns: l=3 -> left halo (gw = w0-1); l=4..67 -> body (gw = w0..w0+63,
// 16B-aligned for b128 staging); l=68 -> right halo (gw = w0+64).

typedef float v4f __attribute__((ext_vector_type(4)));
typedef int   v4i __attribute__((ext_vector_type(4)));
typedef __attribute__((address_space(1))) int gint_t;   // global int
typedef __attribute__((address_space(3))) int lint_t;   // LDS int
typedef __attribute__((address_space(1))) v4i gv4i_t;   // global int4
typedef __attribute__((address_space(3))) v4i lv4i_t;   // LDS int4

#if __has_builtin(__builtin_amdgcn_global_load_async_to_lds_b128)
#define HAVE_ASYNC_B128 1
#else
#define HAVE_ASYNC_B128 0
#endif
#if __has_builtin(__builtin_amdgcn_global_load_async_to_lds_b32)
#define HAVE_ASYNC_B32 1
#else
#define HAVE_ASYNC_B32 0
#endif

__global__ __launch_bounds__(256) void sobel_cat16_kernel(
    const float* __restrict__ x, float* __restrict__ out)
{
    __shared__ float tile[LDS_H * LDS_W];

    const int tid = threadIdx.x;

    // Decode: blockIdx.x = slice*64 + tileY*4 + tileX
    const int blk   = blockIdx.x;
    const int s_idx = blk >> 6;            // 0..127  (= b*64 + d)
    const int tl    = blk & 63;
    const int h0    = (tl >> 2) * TILE_H;  // tile origin row
    const int w0    = (tl & 3) * TILE_W;   // tile origin col
    const int b     = s_idx >> 6;
    const int d     = s_idx & 63;

    const float* gslice = x + (size_t)s_idx * IMHW;

    // ---- Stage halo tile into LDS via CDNA5 async global->LDS (ASYNCcnt).
    // Body: 18 rows x 16 float4 = 288 b128 tasks (rows OOB only at image
    // top/bottom -> zero-filled). Edges: 2 cols x 18 rows b32 tasks.
    // OOB slots are the exact complement of async writes -> no LDS race,
    // and zero-fill matches the conv's SAME zero padding.
#pragma unroll 2
    for (int j = tid; j < LDS_H * 16; j += 256) {
        const int tr  = j >> 4;            // 0..17
        const int qc  = j & 15;            // 0..15 -> 4 floats each
        const int gh  = h0 - 1 + tr;
        float* lp = &tile[tr * LDS_W + 4 + 4 * qc];
        if ((gh >= 0) & (gh < IMH)) {
            const float* gp = gslice + gh * IMW + w0 + 4 * qc;
#if HAVE_ASYNC_B128
            __builtin_amdgcn_global_load_async_to_lds_b128(
                (gv4i_t*)gp, (lv4i_t*)lp, 0, 0);
#elif HAVE_ASYNC_B32
            __builtin_amdgcn_global_load_async_to_lds_b32((gint_t*)(gp + 0), (lint_t*)(lp + 0), 0, 0);
            __builtin_amdgcn_global_load_async_to_lds_b32((gint_t*)(gp + 1), (lint_t*)(lp + 1), 0, 0);
            __builtin_amdgcn_global_load_async_to_lds_b32((gint_t*)(gp + 2), (lint_t*)(lp + 2), 0, 0);
            __builtin_amdgcn_global_load_async_to_lds_b32((gint_t*)(gp + 3), (lint_t*)(lp + 3), 0, 0);
#else
            *(v4f*)lp = *(const v4f*)gp;
#endif
        } else {
            *(v4f*)lp = (v4f)0.0f;
        }
    }
    if (tid < 2 * LDS_H) {                 // edge columns
        const int left = tid < LDS_H;
        const int tr   = left ? tid : tid - LDS_H;
        const int gh   = h0 - 1 + tr;
        const int gw   = left ? (w0 - 1) : (w0 + TILE_W);
        float* lp = &tile[tr * LDS_W + (left ? 3 : 68)];
        if (((gh >= 0) & (gh < IMH)) && ((gw >= 0) & (gw < IMW))) {
            const float* gp = gslice + gh * IMW + gw;
#if HAVE_ASYNC_B32
            __builtin_amdgcn_global_load_async_to_lds_b32(
                (gint_t*)gp, (lint_t*)lp, 0, 0);
#else
            *lp = *gp;
#endif
        } else {
            *lp = 0.0f;
        }
    }
#if HAVE_ASYNC_B128 || HAVE_ASYNC_B32
#if __has_builtin(__builtin_amdgcn_s_wait_asynccnt)
    __builtin_amdgcn_s_wait_asynccnt(0);
#else
    asm volatile("s_wait_asynccnt 0" ::: "memory");
#endif
#endif
    __syncthreads();

    // ---- Compute: thread (r, q) -> output row h0+r, cols w0+4q .. +3
    const int r  = tid >> 4;          // 0..15
    const int c0 = (tid & 15) * 4;    // 0,4,...,60

    const float* row0 = &tile[(r + 0) * LDS_W + c0 + 3];
    const float* row1 = &tile[(r + 1) * LDS_W + c0 + 3];
    const float* row2 = &tile[(r + 2) * LDS_W + c0 + 3];

    // Separable Sobel: vertical smooth sm[j] = a+2b+c, vertical diff tv[j] = c-a
    float sm[6], tv[6], xv[4];
#pragma unroll
    for (int j = 0; j < 6; ++j) {
        const float a = row0[j];
        const float m = row1[j];
        const float c = row2[j];
        sm[j] = a + 2.0f * m + c;
        tv[j] = c - a;
        if (j >= 1 && j <= 4) xv[j - 1] = m;   // center pixels (the x copy)
    }

    v4f mag4, xv4;
#pragma unroll
    for (int c = 0; c < 4; ++c) {
        const float gx = sm[c + 2] - sm[c];
        const float gy = tv[c] + 2.0f * tv[c + 1] + tv[c + 2];
        mag4[c] = __builtin_sqrtf(gx * gx + gy * gy);
        xv4[c]  = xv[c];
    }

    // ---- 16 coalesced 128-bit non-temporal stores (output is streaming,
    // 512 MiB > 192 MB L2, never re-read).
    const int h = h0 + r;
    const size_t base0 =
        ((size_t)(b * 1024 + d)) * IMHW + (size_t)h * IMW + (size_t)(w0 + c0);
#pragma unroll
    for (int k = 0; k < 8; ++k) {
        v4f* p_s = (v4f*)(out + base0 + (size_t)(2 * k) * CH_STRIDE);
        v4f* p_x = (v4f*)(out + base0 + (size_t)(2 * k + 1) * CH_STRIDE);
        __builtin_nontemporal_store(mag4, p_s);
        __builtin_nontemporal_store(xv4, p_x);
    }
}

extern "C" void kernel_launch(void* const* d_in, const int* in_sizes, int n_in,
                              void* d_out, int out_size, void* d_ws, size_t ws_size,
                              hipStream_t stream)
{
    (void)in_sizes; (void)n_in; (void)d_ws; (void)ws_size; (void)out_size;
    const float* xin = (const float*)d_in[0];
    float* outp = (float*)d_out;
    // 128 slices * 64 tiles per slice
    dim3 grid(NSLICE * 64), block(256);
    hipLaunchKernelGGL(sobel_cat16_kernel, grid, block, 0, stream, xin, outp);
}